// GraphNet_42236708389589
// MI455X (gfx1250) — compile-verified
//
#include <hip/hip_runtime.h>
#include <hip/hip_bf16.h>

// ---------------------------------------------------------------------------
// GraphNet forward on gfx1250 (MI455X).
//   edges [E,K=5,D=8] --MLP(8->32->relu->6)--> scatter-mean by receiver
//   channels [N,K,8]  --MLP(8->32->relu->6)--> ch_emb
//   nodes = mean_k(messages + ch_emb); actor 6->32->3; critic 6->16->1
// Edge/channel MLPs run on V_WMMA_F32_16X16X4_F32 (16-row tiles per wave32).
// HBM-bound: 256 MB edge stream dominates; WMMA keeps compute off the
// critical path. All layer-2 weight loads are branch-free (clamped index +
// mask-multiply) so no EXEC manipulation occurs between WMMAs.
// ---------------------------------------------------------------------------

typedef float v2f __attribute__((ext_vector_type(2)));
typedef float v8f __attribute__((ext_vector_type(8)));

__device__ __forceinline__ v8f wmma_f32_16x16x4(v2f a, v2f b, v8f c) {
  // 8-arg pattern: (neg_a, A, neg_b, B, c_mod, C, reuse_a, reuse_b)
  return __builtin_amdgcn_wmma_f32_16x16x4_f32(false, a, false, b, (short)0, c,
                                               false, false);
}

__device__ __forceinline__ float relu(float x) { return x > 0.0f ? x : 0.0f; }

// Scalar fallback for partial tiles: one row of MLP(8 -> 32 -> relu -> 6).
__device__ void mlp_row_scalar(const float* __restrict__ x,
                               const float* __restrict__ W1,
                               const float* __restrict__ b1,
                               const float* __restrict__ W2,
                               const float* __restrict__ b2,
                               float* __restrict__ out6) {
  float h[32];
#pragma unroll
  for (int j = 0; j < 32; ++j) {
    float s = b1[j];
#pragma unroll
    for (int d = 0; d < 8; ++d) s += x[d] * W1[d * 32 + j];
    h[j] = relu(s);
  }
#pragma unroll
  for (int c = 0; c < 6; ++c) {
    float s = b2[c];
#pragma unroll
    for (int j = 0; j < 32; ++j) s += h[j] * W2[j * 6 + c];
    out6[c] = s;
  }
}

// One wave computes MLP(8->32->relu->6) for 16 rows using WMMA f32 16x16x4.
// SCATTER=true : atomically add results into sums[(recv*5+kch)*6 + c]
// SCATTER=false: store results to out[row*6 + c]
template <bool SCATTER>
__global__ __launch_bounds__(256) void mlp_tile_kernel(
    const float* __restrict__ x,    // [R, 8] row-major
    const float* __restrict__ W1,   // [8, 32]
    const float* __restrict__ b1,   // [32]
    const float* __restrict__ W2,   // [32, 6]
    const float* __restrict__ b2,   // [6]
    const int* __restrict__ receivers,  // [E] (used when SCATTER)
    float* __restrict__ out,
    long long R) {
  __shared__ float lds_h[8][16 * 32];  // per-wave 16x32 activation staging

  const int wave = threadIdx.x >> 5;
  const int lane = threadIdx.x & 31;
  const long long tile = (long long)blockIdx.x * 8 + wave;
  const long long base = tile * 16;
  if (base >= R) return;

  const int lg = lane >> 4;  // lane group (0: lanes 0-15, 1: lanes 16-31)
  const int m  = lane & 15;  // A-row index / B-and-C column index

  if (base + 16 <= R) {
    // ---------------- full 16-row WMMA tile (EXEC uniform) ----------------
    const float* xb = x + base * 8;
    // Prefetch next tile's rows (speculative; OOB is dropped silently).
    __builtin_prefetch(xb + 16 * 8 + m * 8, 0, 1);

    // A-matrix 16x4 f32 layout: lane m holds row m; vgpr j carries
    // K = 2*lg + j (lg=0 -> K0,K1 ; lg=1 -> K2,K3). Two K-chunks cover D=8.
    const float* px = xb + m * 8 + 2 * lg;
    v2f a0 = {px[0], px[1]};  // K-chunk 0 (k = 0..3)
    v2f a1 = {px[4], px[5]};  // K-chunk 1 (k = 4..7)

    // B-matrix 4x16 layout mirrors A: lane n holds col n; vgpr j is
    // K = 2lg + j. Two N-tiles (cols 0-15, 16-31) x two K-chunks.
    const int k0 = 2 * lg;
    v2f b00 = {W1[(k0 + 0) * 32 + m],      W1[(k0 + 1) * 32 + m]};
    v2f b01 = {W1[(k0 + 4) * 32 + m],      W1[(k0 + 5) * 32 + m]};
    v2f b10 = {W1[(k0 + 0) * 32 + 16 + m], W1[(k0 + 1) * 32 + 16 + m]};
    v2f b11 = {W1[(k0 + 4) * 32 + 16 + m], W1[(k0 + 5) * 32 + 16 + m]};

    // C/D 16x16 f32 layout: vgpr g holds row g + 8*lg, column = lane&15.
    const float bb0 = b1[m];
    const float bb1 = b1[16 + m];
    v8f c0 = {bb0, bb0, bb0, bb0, bb0, bb0, bb0, bb0};
    v8f c1 = {bb1, bb1, bb1, bb1, bb1, bb1, bb1, bb1};

    c0 = wmma_f32_16x16x4(a0, b00, c0);
    c0 = wmma_f32_16x16x4(a1, b01, c0);
    c1 = wmma_f32_16x16x4(a0, b10, c1);
    c1 = wmma_f32_16x16x4(a1, b11, c1);

    // ReLU; stage H (16x32) row-major in LDS to re-stripe C-layout -> A-layout.
    float* lh = lds_h[wave];
#pragma unroll
    for (int g = 0; g < 8; ++g) {
      lh[(g + 8 * lg) * 32 + m]      = relu(c0[g]);
      lh[(g + 8 * lg) * 32 + 16 + m] = relu(c1[g]);
    }
    // Same-wave DS RAW: compiler inserts s_wait_dscnt; no barrier needed.

    // Layer 2: [16x32] x [32x6] with N padded to 16 (cols >=6 are zero).
    // Branch-free padding: clamp column index to a valid address and zero
    // the value with a 0/1 mask so no EXEC-divergent loads sit between
    // WMMAs (WMMA requires EXEC all-1s and benefits from load clustering).
    const bool act = (m < 6);
    const float msk = act ? 1.0f : 0.0f;
    const int mc = act ? m : 0;  // clamped, always-valid column
    const float cb = b2[mc] * msk;
    v8f c2 = {cb, cb, cb, cb, cb, cb, cb, cb};
#pragma unroll
    for (int kc = 0; kc < 8; ++kc) {
      const int k = 4 * kc + 2 * lg;
      v2f av = {lh[m * 32 + k], lh[m * 32 + k + 1]};
      v2f bv = {W2[(k + 0) * 6 + mc] * msk, W2[(k + 1) * 6 + mc] * msk};
      c2 = wmma_f32_16x16x4(av, bv, c2);
    }

    // Emit: lane with column m<6 owns rows g+8*lg of the 16x6 result.
    if (act) {
#pragma unroll
      for (int g = 0; g < 8; ++g) {
        const long long r = base + g + 8 * lg;
        const float v = c2[g];
        if (SCATTER) {
          const long long e = r / 5;
          const int kch = (int)(r - e * 5);
          const int recv = receivers[e];
          unsafeAtomicAdd(&out[((long long)recv * 5 + kch) * 6 + m], v);
        } else {
          out[r * 6 + m] = v;
        }
      }
    }
  } else {
    // ---------------- tail tile: scalar per-lane fallback ----------------
    const long long r = base + lane;
    if (r < R) {
      float o[6];
      mlp_row_scalar(x + r * 8, W1, b1, W2, b2, o);
      if (SCATTER) {
        const long long e = r / 5;
        const int kch = (int)(r - e * 5);
        const int recv = receivers[e];
#pragma unroll
        for (int c = 0; c < 6; ++c)
          unsafeAtomicAdd(&out[((long long)recv * 5 + kch) * 6 + c], o[c]);
      } else {
#pragma unroll
        for (int c = 0; c < 6; ++c) out[r * 6 + c] = o[c];
      }
    }
  }
}

__global__ __launch_bounds__(256) void count_edges_kernel(
    const int* __restrict__ receivers, float* __restrict__ counts, int E) {
  const int e = blockIdx.x * blockDim.x + threadIdx.x;
  if (e < E) unsafeAtomicAdd(&counts[receivers[e]], 1.0f);
}

__global__ __launch_bounds__(256) void finalize_kernel(
    const float* __restrict__ sums,    // [N,5,6]
    const float* __restrict__ counts,  // [N]
    const float* __restrict__ chemb,   // [N,5,6]
    const float* __restrict__ Wa1, const float* __restrict__ ba1,  // [6,32],[32]
    const float* __restrict__ Wa2, const float* __restrict__ ba2,  // [32,3],[3]
    const float* __restrict__ Wv1, const float* __restrict__ bv1,  // [6,16],[16]
    const float* __restrict__ Wv2, const float* __restrict__ bv2,  // [16,1],[1]
    float* __restrict__ out,  // [N*3] logits ++ [N] value
    int N) {
  const int i = blockIdx.x * blockDim.x + threadIdx.x;
  if (i >= N) return;

  float cnt = counts[i];
  cnt = cnt > 1.0f ? cnt : 1.0f;
  const float inv = 1.0f / cnt;

  float nodes[6] = {0, 0, 0, 0, 0, 0};
  const long long b = (long long)i * 30;
#pragma unroll
  for (int k = 0; k < 5; ++k)
#pragma unroll
    for (int c = 0; c < 6; ++c)
      nodes[c] += sums[b + k * 6 + c] * inv + chemb[b + k * 6 + c];
#pragma unroll
  for (int c = 0; c < 6; ++c) nodes[c] *= 0.2f;

  // actor: 6 -> 32 -> relu -> 3
  float h[32];
#pragma unroll
  for (int j = 0; j < 32; ++j) {
    float s = ba1[j];
#pragma unroll
    for (int c = 0; c < 6; ++c) s += nodes[c] * Wa1[c * 32 + j];
    h[j] = relu(s);
  }
#pragma unroll
  for (int mo = 0; mo < 3; ++mo) {
    float s = ba2[mo];
#pragma unroll
    for (int j = 0; j < 32; ++j) s += h[j] * Wa2[j * 3 + mo];
    out[(long long)i * 3 + mo] = s;
  }

  // critic: 6 -> 16 -> relu -> 1
  float hv[16];
#pragma unroll
  for (int j = 0; j < 16; ++j) {
    float s = bv1[j];
#pragma unroll
    for (int c = 0; c < 6; ++c) s += nodes[c] * Wv1[c * 16 + j];
    hv[j] = relu(s);
  }
  float v = bv2[0];
#pragma unroll
  for (int j = 0; j < 16; ++j) v += hv[j] * Wv2[j];
  out[(long long)N * 3 + i] = v;
}

extern "C" void kernel_launch(void* const* d_in, const int* in_sizes, int n_in,
                              void* d_out, int out_size, void* d_ws,
                              size_t ws_size, hipStream_t stream) {
  const float* edges    = (const float*)d_in[0];
  const float* channels = (const float*)d_in[1];
  const float* We1 = (const float*)d_in[2];
  const float* be1 = (const float*)d_in[3];
  const float* We2 = (const float*)d_in[4];
  const float* be2 = (const float*)d_in[5];
  const float* Wc1 = (const float*)d_in[6];
  const float* bc1 = (const float*)d_in[7];
  const float* Wc2 = (const float*)d_in[8];
  const float* bc2 = (const float*)d_in[9];
  const float* Wa1 = (const float*)d_in[10];
  const float* ba1 = (const float*)d_in[11];
  const float* Wa2 = (const float*)d_in[12];
  const float* ba2 = (const float*)d_in[13];
  const float* Wv1 = (const float*)d_in[14];
  const float* bv1 = (const float*)d_in[15];
  const float* Wv2 = (const float*)d_in[16];
  const float* bv2 = (const float*)d_in[17];
  const int* receivers = (const int*)d_in[18];

  const int K = 5, D = 8;
  const int E = in_sizes[18];
  const int N = in_sizes[1] / (K * D);

  // Workspace layout: sums[N*K*6] | counts[N] | chemb[N*K*6]  (~12.2 MB)
  float* sums   = (float*)d_ws;
  float* counts = sums + (size_t)N * K * 6;
  float* chemb  = counts + N;
  const size_t zero_bytes = ((size_t)N * K * 6 + N) * sizeof(float);
  hipMemsetAsync(d_ws, 0, zero_bytes, stream);

  // 1) per-node edge counts
  count_edges_kernel<<<(E + 255) / 256, 256, 0, stream>>>(receivers, counts, E);

  // 2) edge MLP + scatter-add (WMMA): R = E*K rows of 8 floats
  {
    const long long R = (long long)E * K;
    const long long tiles = (R + 15) / 16;
    const int blocks = (int)((tiles + 7) / 8);
    mlp_tile_kernel<true><<<blocks, 256, 0, stream>>>(
        edges, We1, be1, We2, be2, receivers, sums, R);
  }

  // 3) channel MLP (WMMA), dense store of ch_emb
  {
    const long long R = (long long)N * K;
    const long long tiles = (R + 15) / 16;
    const int blocks = (int)((tiles + 7) / 8);
    mlp_tile_kernel<false><<<blocks, 256, 0, stream>>>(
        channels, Wc1, bc1, Wc2, bc2, receivers, chemb, R);
  }

  // 4) per-node mean + actor/critic heads
  finalize_kernel<<<(N + 255) / 256, 256, 0, stream>>>(
      sums, counts, chemb, Wa1, ba1, Wa2, ba2, Wv1, bv1, Wv2, bv2,
      (float*)d_out, N);
}